// AdditiveAttention_37074157699814
// MI455X (gfx1250) — compile-verified
//
#include <hip/hip_runtime.h>
#include <math.h>

// Problem constants (from the reference).
constexpr int BB = 4, NQ = 256, NK = 2048, DD = 64, HH = 32, VD = 64;
constexpr int QT = 16;   // query tile (WMMA M)
constexpr int KB = 32;   // key block per iteration (8 WMMA K-steps of 4)

typedef __attribute__((ext_vector_type(2))) float v2f;
typedef __attribute__((ext_vector_type(8))) float v8f;

__device__ __forceinline__ float fast_tanh(float x) {
#if defined(__gfx1250__) && __has_builtin(__builtin_amdgcn_tanhf)
    return __builtin_amdgcn_tanhf(x);   // v_tanh_f32 (CDNA5 TRANS op)
#else
    return tanhf(x);
#endif
}

// ---------------------------------------------------------------------------
// Kernel 1: projections  out[r][h] = sum_d X[r][d] * W[d][h]   (H = 32, D = 64)
// ---------------------------------------------------------------------------
__global__ __launch_bounds__(256) void addattn_proj_kernel(
    const float* __restrict__ X, const float* __restrict__ W,
    float* __restrict__ out, int rows)
{
    int gid = blockIdx.x * blockDim.x + threadIdx.x;
    if (gid >= rows * HH) return;
    int r = gid >> 5;
    int h = gid & 31;
    const float* xr = X + (size_t)r * DD;
    float acc = 0.f;
#pragma unroll
    for (int d = 0; d < DD; ++d)
        acc = fmaf(xr[d], W[d * HH + h], acc);
    out[gid] = acc;
}

// ---------------------------------------------------------------------------
// Kernel 2: fused  tanh-score -> mask -> online softmax -> attn @ V  (WMMA f32)
// Grid: BB * (NQ/QT) blocks, 128 threads (4 waves of 32).
// Wave w owns V_DIM columns [16w, 16w+16).
// ---------------------------------------------------------------------------
__global__ __launch_bounds__(128) void addattn_fused_kernel(
    const float* __restrict__ qp,          // [BB*NQ][HH]
    const float* __restrict__ kp,          // [BB*NK][HH]
    const float* __restrict__ values,      // [BB][NK][VD]
    const int*   __restrict__ valid_lens,  // [BB]
    const float* __restrict__ wv,          // [HH]
    float*       __restrict__ out)         // [BB][NQ][VD]
{
    __shared__ float qp_l[QT * 33];        // query projections, padded stride
    __shared__ float wv_l[HH];
    __shared__ float s_l[QT * 33];         // scores -> probabilities (in place)
    __shared__ float v_l[KB * VD];         // V block
    __shared__ float m_l[QT], l_l[QT], sc_l[QT];

    const int tid  = threadIdx.x;
    const int bq   = blockIdx.x;
    const int b    = bq >> 4;              // NQ/QT = 16 tiles per batch
    const int q0   = (bq & 15) * QT;
    const int vlen = valid_lens[b];

    // --- preamble: stage query tile + wv, init online-softmax state -------
    if (tid < HH) wv_l[tid] = wv[tid];
    {
        int e = tid * 4;                   // 128 threads * 4 = 512 = QT*HH
#pragma unroll
        for (int j = 0; j < 4; ++j) {
            int r = (e + j) >> 5, h = (e + j) & 31;
            qp_l[r * 33 + h] = qp[((size_t)b * NQ + q0 + r) * HH + h];
        }
    }
    if (tid < QT) { m_l[tid] = -3.0e38f; l_l[tid] = 0.f; }

    const int lane = tid & 31;
    const int wid  = tid >> 5;             // wave id, uniform per wave
    const int half = lane >> 4;
    const int lm   = lane & 15;
    const int n0   = wid * 16;             // this wave's V_DIM slice
    const int c    = lane;                 // score column handled by this thread

    v8f acc = {0.f, 0.f, 0.f, 0.f, 0.f, 0.f, 0.f, 0.f};

    for (int kb = 0; kb < NK; kb += KB) {
        __syncthreads();                   // protect v_l / s_l from prev iter

        // --- stage V block (coalesced float4) -----------------------------
        {
            int row = tid >> 2, seg = tid & 3;
            const float4* vg =
                (const float4*)(values + ((size_t)b * NK + kb + row) * VD + seg * 16);
            float4* vl = (float4*)(&v_l[row * VD + seg * 16]);
            vl[0] = vg[0]; vl[1] = vg[1]; vl[2] = vg[2]; vl[3] = vg[3];
        }

        // --- scores: this thread owns column c, rows wid + 4i -------------
        float kpr[HH];
        {
            const float4* kg = (const float4*)(kp + ((size_t)b * NK + kb + c) * HH);
#pragma unroll
            for (int j = 0; j < 8; ++j) {
                float4 t4 = kg[j];
                kpr[4*j] = t4.x; kpr[4*j+1] = t4.y; kpr[4*j+2] = t4.z; kpr[4*j+3] = t4.w;
            }
        }
        const bool valid = (kb + c) < vlen;
#pragma unroll
        for (int i = 0; i < 4; ++i) {
            int r = wid + 4 * i;
            float s = 0.f;
#pragma unroll
            for (int h = 0; h < HH; ++h)
                s = fmaf(wv_l[h], fast_tanh(qp_l[r * 33 + h] + kpr[h]), s);
            s_l[r * 33 + c] = valid ? s : -1.0e6f;
        }
        __syncthreads();

        // --- online softmax stats: 16 threads, one row each ---------------
        if (tid < QT) {
            const int row = tid;
            float m_old = m_l[row];
            float mb = -3.0e38f;
            for (int cc = 0; cc < KB; ++cc) mb = fmaxf(mb, s_l[row * 33 + cc]);
            float m_new = fmaxf(m_old, mb);
            float scale = __expf(m_old - m_new);   // underflows to 0 on first block
            float sum = 0.f;
            for (int cc = 0; cc < KB; ++cc) {
                float p = __expf(s_l[row * 33 + cc] - m_new);
                s_l[row * 33 + cc] = p;            // scores -> probs in place
                sum += p;
            }
            m_l[row]  = m_new;
            l_l[row]  = l_l[row] * scale + sum;
            sc_l[row] = scale;
        }
        __syncthreads();

        // --- rescale accumulator, then 8 x V_WMMA_F32_16X16X4_F32 ---------
        // C/D layout: VGPR r -> row (r + 8*half), col lm.
#pragma unroll
        for (int r = 0; r < 8; ++r) acc[r] *= sc_l[r + 8 * half];

        // A/B layout (32-bit, 16x4 / 4x16): register j, half h -> K = j + 2h.
#pragma unroll
        for (int t = 0; t < 8; ++t) {
            int kA = 4 * t + 2 * half;
            v2f a  = { s_l[lm * 33 + kA],      s_l[lm * 33 + kA + 1] };
            v2f bb = { v_l[kA * VD + n0 + lm], v_l[(kA + 1) * VD + n0 + lm] };
            acc = __builtin_amdgcn_wmma_f32_16x16x4_f32(
                false, a, false, bb, (short)0, acc, false, false);
        }
    }

    // --- epilogue: divide by l and store --------------------------------
#pragma unroll
    for (int r = 0; r < 8; ++r) {
        int q = r + 8 * half;
        out[((size_t)b * NQ + q0 + q) * VD + n0 + lm] = acc[r] / l_l[q];
    }
}

// ---------------------------------------------------------------------------
extern "C" void kernel_launch(void* const* d_in, const int* in_sizes, int n_in,
                              void* d_out, int out_size, void* d_ws, size_t ws_size,
                              hipStream_t stream) {
    (void)in_sizes; (void)n_in; (void)out_size; (void)ws_size;
    const float* queries    = (const float*)d_in[0];
    const float* keys       = (const float*)d_in[1];
    const float* values     = (const float*)d_in[2];
    const int*   valid_lens = (const int*)  d_in[3];
    const float* Wq         = (const float*)d_in[4];
    const float* Wk         = (const float*)d_in[5];
    const float* wv         = (const float*)d_in[6];
    float* out = (float*)d_out;

    float* qp = (float*)d_ws;                       // BB*NQ*HH = 32K floats
    float* kp = qp + (size_t)BB * NQ * HH;          // BB*NK*HH = 256K floats

    addattn_proj_kernel<<<(BB * NQ * HH + 255) / 256, 256, 0, stream>>>(
        queries, Wq, qp, BB * NQ);
    addattn_proj_kernel<<<(BB * NK * HH + 255) / 256, 256, 0, stream>>>(
        keys, Wk, kp, BB * NK);
    addattn_fused_kernel<<<BB * (NQ / QT), 128, 0, stream>>>(
        qp, kp, values, valid_lens, wv, out);
}